// NodeNetGNN_52226802319462
// MI455X (gfx1250) — compile-verified
//
#include <hip/hip_runtime.h>

// ---------------------------------------------------------------------------
// NodeNetGNN for MI455X (gfx1250, wave32).
//
// Pipeline (all on `stream`, in dependency order):
//   Z0  zero deg_node/deg_net/agg (contiguous ws region) and node_out region
//   K1  degree histogram (atomic f32 adds of 1.0 -> exact)
//   K2  transpose W2 [16,1024] -> W2t [32,512]  (W2t[i,p*32+o] = W2[p,i*32+o])
//   K3  netb[net,o] = sum_i net_feat[net,i] * b2[i*32+o]
//   G1  WMMA GEMM: Mbuf[20000,512] = net_feat[20000,32] @ W2t[32,512]
//   K4  scatter: agg[net,f] += node_feat[node,f] * rsqrt(max(deg_node,1))
//   G2  WMMA GEMM: net_out = (agg * rsqrt(max(deg_net,1))) @ W1 + b1
//   K5  per-edge: msg[o] = netb[net,o] + sum_p pin[e,p]*Mbuf[net,p*32+o];
//       atomicAdd into node_out accumulator
//   K6  finalize: node_out = acc / max(deg_node,1) + b_nn
// ---------------------------------------------------------------------------

typedef __attribute__((ext_vector_type(2))) float v2f;
typedef __attribute__((ext_vector_type(8))) float v8f;

enum : int {
  kNNode = 100000, kNNet = 20000, kE = 200000,
  kHNode = 64, kHNet = 32, kHPin = 16, kONode = 32, kONet = 64,
  kMCols = kHPin * kONode   // 512 = inner width of Mbuf / W2t
};

__global__ void zero_f32(float* __restrict__ p, int n) {
  int t = blockIdx.x * blockDim.x + threadIdx.x;
  if (t < n) p[t] = 0.0f;
}

__global__ void degree_kernel(const int* __restrict__ edge_node,
                              const int* __restrict__ edge_net,
                              float* __restrict__ deg_node,
                              float* __restrict__ deg_net, int e_cnt) {
  int e = blockIdx.x * blockDim.x + threadIdx.x;
  if (e >= e_cnt) return;
  atomicAdd(&deg_node[edge_node[e]], 1.0f);
  atomicAdd(&deg_net[edge_net[e]], 1.0f);
}

// W2t[i*512 + p*32 + o] = W2[p*1024 + i*32 + o]
__global__ void transpose_w2(const float* __restrict__ W2, float* __restrict__ W2t) {
  int t = blockIdx.x * blockDim.x + threadIdx.x;   // 16384 dests
  if (t >= kHNet * kMCols) return;
  int i = t >> 9;              // 0..31
  int p = (t >> 5) & 15;       // 0..15
  int o = t & 31;              // 0..31
  W2t[t] = W2[p * (kHNet * kONode) + i * kONode + o];
}

// netb[net*32+o] = sum_i net_feat[net*32+i] * b2[i*32+o]
__global__ void netb_kernel(const float* __restrict__ net_feat,
                            const float* __restrict__ b2,
                            float* __restrict__ netb, int n) {
  int t = blockIdx.x * blockDim.x + threadIdx.x;
  if (t >= n) return;
  int net = t >> 5, o = t & 31;
  const float* nf = net_feat + (size_t)net * kHNet;
  float acc = 0.0f;
#pragma unroll
  for (int i = 0; i < kHNet; ++i) acc = fmaf(nf[i], b2[i * kONode + o], acc);
  netb[t] = acc;
}

// agg[net*64+f] += node_feat[node*64+f] * rsqrt(max(deg_node[node],1))
__global__ void scatter_agg(const float* __restrict__ node_feat,
                            const int* __restrict__ edge_node,
                            const int* __restrict__ edge_net,
                            const float* __restrict__ deg_node,
                            float* __restrict__ agg, int e_cnt) {
  int t = blockIdx.x * blockDim.x + threadIdx.x;
  if (t >= e_cnt * kHNode) return;
  int e = t >> 6, f = t & 63;
  int node = edge_node[e], net = edge_net[e];
  float s = __frsqrt_rn(fmaxf(deg_node[node], 1.0f));
  atomicAdd(&agg[(size_t)net * kHNode + f], node_feat[(size_t)node * kHNode + f] * s);
}

// ---------------------------------------------------------------------------
// f32 WMMA GEMM: C[M,N] = (rowscale(A))[M,K] @ B[K,N] (+ bias), row-major.
// One wave computes one 16x16 tile via V_WMMA_F32_16X16X4_F32 (K-step 4).
// ISA layouts (wave32):
//   A 16x4:  lane L holds A[m=L&15, k0], A[m, k0+1],  k0 = 2*(L>>4)
//   B 4x16:  lane L holds B[k0, n=L&15], B[k0+1, n]
//   C 16x16: VGPR r of lane L holds C[r + 8*(L>>4), L&15]
// degA != nullptr -> scale A row by rsqrt(max(degA[row],1)).
// Grid is sized so every wave owns a full tile (EXEC all-1s at each WMMA).
// ---------------------------------------------------------------------------
__global__ __launch_bounds__(256)
void gemm_wmma_f32(const float* __restrict__ A, const float* __restrict__ B,
                   float* __restrict__ C, int M, int N, int K,
                   const float* __restrict__ degA, const float* __restrict__ bias) {
  int wave = blockIdx.x * (blockDim.x >> 5) + (threadIdx.x >> 5);
  int lane = threadIdx.x & 31;
  int ntiles = N >> 4;
  int total = (M >> 4) * ntiles;
  if (wave >= total) return;                 // wave-uniform; full waves only
  int tm = wave / ntiles, tn = wave % ntiles;
  int half = lane >> 4;                      // 0 | 1
  int l15  = lane & 15;
  int arow = tm * 16 + l15;                  // this lane's A/M row
  int col  = tn * 16 + l15;                  // this lane's B/C column
  float scale = 1.0f;
  if (degA) scale = __frsqrt_rn(fmaxf(degA[arow], 1.0f));

  const float* Ar = A + (size_t)arow * K;
  v8f acc = {};
  for (int k = 0; k < K; k += 4) {
    int ks = k + half * 2;
    v2f a, b;
    a.x = Ar[ks]     * scale;
    a.y = Ar[ks + 1] * scale;
    b.x = B[(size_t)ks * N + col];
    b.y = B[(size_t)(ks + 1) * N + col];
    acc = __builtin_amdgcn_wmma_f32_16x16x4_f32(false, a, false, b,
                                                (short)0, acc, false, false);
  }
  float bv = bias ? bias[col] : 0.0f;
#pragma unroll
  for (int r = 0; r < 8; ++r) {
    int orow = tm * 16 + r + half * 8;
    C[(size_t)orow * N + col] = acc[r] + bv;
  }
}

// One wave per edge; lane = output channel o (32 lanes = O_NODE).
// msg[o] = netb[net,o] + sum_p pin[e,p] * Mbuf[net, p*32+o]; atomic into acc.
__global__ __launch_bounds__(256)
void edge_msg(const float* __restrict__ pin_feat,
              const int* __restrict__ edge_node,
              const int* __restrict__ edge_net,
              const float* __restrict__ Mbuf,
              const float* __restrict__ netb,
              float* __restrict__ node_acc, int e_cnt) {
  int e = blockIdx.x * (blockDim.x >> 5) + (threadIdx.x >> 5);
  int o = threadIdx.x & 31;
  if (e >= e_cnt) return;
  int net = edge_net[e], node = edge_node[e];
  const float* Me = Mbuf + (size_t)net * kMCols;   // L2-resident (41MB << 192MB)
  const float* pe = pin_feat + (size_t)e * kHPin;
  float acc = netb[(size_t)net * kONode + o];
#pragma unroll
  for (int p = 0; p < kHPin; ++p)
    acc = fmaf(pe[p], Me[p * kONode + o], acc);
  atomicAdd(&node_acc[(size_t)node * kONode + o], acc);
}

// node_out = acc / max(deg,1) + b_nn  (in place over the accumulator)
__global__ void finalize_node(float* __restrict__ node_out,
                              const float* __restrict__ deg_node,
                              const float* __restrict__ b_nn, int n) {
  int t = blockIdx.x * blockDim.x + threadIdx.x;
  if (t >= n) return;
  int node = t >> 5, o = t & 31;
  node_out[t] = node_out[t] / fmaxf(deg_node[node], 1.0f) + b_nn[o];
}

extern "C" void kernel_launch(void* const* d_in, const int* in_sizes, int n_in,
                              void* d_out, int out_size, void* d_ws, size_t ws_size,
                              hipStream_t stream) {
  const float* node_feat = (const float*)d_in[0];
  const float* net_feat  = (const float*)d_in[1];
  const float* pin_feat  = (const float*)d_in[2];
  const int*   edge_node = (const int*)  d_in[3];
  const int*   edge_net  = (const int*)  d_in[4];
  const float* W1        = (const float*)d_in[5];
  const float* b1        = (const float*)d_in[6];
  const float* W2        = (const float*)d_in[7];
  const float* b2        = (const float*)d_in[8];
  const float* b_nn      = (const float*)d_in[9];

  float* out      = (float*)d_out;
  float* node_out = out;                                  // [100000,32]
  float* net_out  = out + (size_t)kNNode * kONode;        // [20000,64]

  // Workspace carve-up (floats): 12,296,384 total (~47 MB)
  float* ws       = (float*)d_ws;
  float* deg_node = ws;                                   // 100000
  float* deg_net  = deg_node + kNNode;                    // 20000
  float* agg      = deg_net  + kNNet;                     // 20000*64
  float* W2t      = agg      + (size_t)kNNet * kHNode;    // 32*512
  float* netb     = W2t      + kHNet * kMCols;            // 20000*32
  float* Mbuf     = netb     + (size_t)kNNet * kONode;    // 20000*512

  const int zeroA = kNNode + kNNet + kNNet * kHNode;      // deg_node|deg_net|agg contiguous
  const int nodeOutN = kNNode * kONode;

  zero_f32<<<(zeroA + 255) / 256, 256, 0, stream>>>(deg_node, zeroA);
  zero_f32<<<(nodeOutN + 255) / 256, 256, 0, stream>>>(node_out, nodeOutN);

  degree_kernel<<<(kE + 255) / 256, 256, 0, stream>>>(edge_node, edge_net,
                                                      deg_node, deg_net, kE);

  transpose_w2<<<(kHNet * kMCols) / 256, 256, 0, stream>>>(W2, W2t);

  netb_kernel<<<(kNNet * kONode) / 256, 256, 0, stream>>>(net_feat, b2, netb,
                                                          kNNet * kONode);

  // G1: Mbuf = net_feat @ W2t ; tiles = (20000/16)*(512/16) = 40000 waves
  gemm_wmma_f32<<<40000 / 8, 256, 0, stream>>>(net_feat, W2t, Mbuf,
                                               kNNet, kMCols, kHNet,
                                               nullptr, nullptr);

  scatter_agg<<<(kE * kHNode) / 256, 256, 0, stream>>>(node_feat, edge_node,
                                                       edge_net, deg_node, agg, kE);

  // G2: net_out = (agg * rsqrt(deg_net)) @ W1 + b1 ; 5000 tiles
  gemm_wmma_f32<<<5000 / 8, 256, 0, stream>>>(agg, W1, net_out,
                                              kNNet, kONet, kHNode,
                                              deg_net, b1);

  edge_msg<<<kE / 8, 256, 0, stream>>>(pin_feat, edge_node, edge_net,
                                       Mbuf, netb, node_out, kE);

  finalize_node<<<(nodeOutN + 255) / 256, 256, 0, stream>>>(node_out, deg_node,
                                                            b_nn, nodeOutN);
}